// RecurrentNeuralNetwork_36163624633116
// MI455X (gfx1250) — compile-verified
//
#include <hip/hip_runtime.h>

#define Bn 128
#define Tn 4096
#define In 64
#define Hn 8
#define Rn (Bn * Tn)   // 524288 rows

typedef __attribute__((ext_vector_type(2))) float v2f;
typedef __attribute__((ext_vector_type(8))) float v8f;

static __device__ __forceinline__ v8f wmma4(v2f a, v2f b, v8f c) {
  // D(16x16,f32) = A(16x4,f32) x B(4x16,f32) + C
  return __builtin_amdgcn_wmma_f32_16x16x4_f32(
      /*neg_a=*/false, a, /*neg_b=*/false, b,
      /*c_mod=*/(short)0, c, /*reuse_a=*/false, /*reuse_b=*/false);
}

// -------------------------------------------------------------------------
// Kernel 1: input projection for BOTH directions fused in the N dimension.
// xp[row, n] = sum_k x[row,k] * w[n][k] + bias[n],  n<8 -> fwd, n>=8 -> bwd
// One wave per 16-row tile; 16 chained K=4 WMMAs over K=64.
// -------------------------------------------------------------------------
__global__ void proj_kernel(const float* __restrict__ x,
                            const float* __restrict__ w_ih_f,
                            const float* __restrict__ w_ih_b,
                            const float* __restrict__ b_f,
                            const float* __restrict__ b_b,
                            float* __restrict__ xp) {
  const int lane = threadIdx.x & 31;
  const int wave = blockIdx.x * (blockDim.x >> 5) + (threadIdx.x >> 5);
  const int m  = lane & 15;   // A row / B,C,D column index for this lane
  const int hi = lane >> 4;   // 0: K=2k, 1: K=2k+2 within a K=4 step
  const long row0 = (long)wave * 16;

  // B matrix [64 x 16]: column n<8 = w_ih_f[n][k], n>=8 = w_ih_b[n-8][k]
  const float* wrow = (m < 8) ? (w_ih_f + m * In) : (w_ih_b + (m - 8) * In);
  v2f bw[16];
#pragma unroll
  for (int ks = 0; ks < 16; ++ks) {
    const int k = ks * 4 + hi * 2;
    bw[ks].x = wrow[k];
    bw[ks].y = wrow[k + 1];
  }
  const float bias = (m < 8) ? b_f[m] : b_b[m - 8];
  v8f acc;
#pragma unroll
  for (int r = 0; r < 8; ++r) acc[r] = bias;

  const float* xr = x + row0 * In;
#pragma unroll
  for (int ks = 0; ks < 16; ++ks) {
    const int k = ks * 4 + hi * 2;
    v2f a;
    a.x = xr[m * In + k];
    a.y = xr[m * In + k + 1];
    acc = wmma4(a, bw[ks], acc);
  }
  // D: VGPR r -> (row r, col m) in lanes 0-15, (row r+8, col m) in lanes 16-31
  float* orow = xp + (row0 + hi * 8) * 16 + m;
#pragma unroll
  for (int r = 0; r < 8; ++r) orow[r * 16] = acc[r];
}

// -------------------------------------------------------------------------
// Kernel 2: the sequential recurrence  h_t = relu(xp_t + W_hh @ h_{t-1}).
// Orientation: D[j][batch] = W_hh(16x8, rows 8..15 = 0) x h(8x16) + xp_t.
// One wave per (direction, 16-batch tile): 2 WMMAs + relu per step.
// -------------------------------------------------------------------------
__global__ void scan_kernel(const float* __restrict__ xp,     // [Rn,16]
                            const float* __restrict__ w_hh_f, // [8,8]
                            const float* __restrict__ w_hh_b, // [8,8]
                            float* __restrict__ hbuf) {       // [Rn,16]
  const int lane = threadIdx.x & 31;
  const int dir  = blockIdx.x >> 3;   // 0 fwd, 1 bwd
  const int bt   = blockIdx.x & 7;    // batch tile (16 batches)
  const int m  = lane & 15;
  const int hi = lane >> 4;
  const float* W = dir ? w_hh_b : w_hh_f;

  // A = W_hh in 16x4 A-layout, rows 8..15 zero, two K=4 steps (K=8 total)
  v2f a0 = {0.f, 0.f}, a1 = {0.f, 0.f};
  if (m < 8) {
    a0.x = W[m * Hn + hi * 2];       a0.y = W[m * Hn + hi * 2 + 1];
    a1.x = W[m * Hn + 4 + hi * 2];   a1.y = W[m * Hn + 4 + hi * 2 + 1];
  }

  v8f h;
#pragma unroll
  for (int r = 0; r < 8; ++r) h[r] = 0.f;

  // lanes 0-15 each own batch (bt*16 + m) for xp loads / h stores
  const long base = ((long)(bt * 16 + m) * Tn) * 16 + dir * 8;
  const int tstart = dir ? (Tn - 1) : 0;
  const int tstep  = dir ? -1 : 1;

  // software pipeline: preload first xp tile
  float4 xa = {0.f, 0.f, 0.f, 0.f}, xb = {0.f, 0.f, 0.f, 0.f};
  if (hi == 0) {
    const float* p = xp + base + (long)tstart * 16;
    xa = *(const float4*)(p);
    xb = *(const float4*)(p + 4);
  }

#pragma unroll 1
  for (int i = 0; i < Tn; ++i) {
    const int t = tstart + i * tstep;

    // C = xp_t tile (rows 8..15 = 0 via hi lanes holding zeros)
    v8f c;
    c[0] = xa.x; c[1] = xa.y; c[2] = xa.z; c[3] = xa.w;
    c[4] = xb.x; c[5] = xb.y; c[6] = xb.z; c[7] = xb.w;

    // prefetch next timestep while WMMAs run
    float4 nxa = {0.f, 0.f, 0.f, 0.f}, nxb = {0.f, 0.f, 0.f, 0.f};
    if (hi == 0 && i + 1 < Tn) {
      const float* p = xp + base + (long)(t + tstep) * 16;
      nxa = *(const float4*)(p);
      nxb = *(const float4*)(p + 4);
    }

    // Build B (8x16) from previous h (D layout -> B layout):
    // lanes 16-31 need rows 2,3 / 6,7 fetched from lanes (lane-16).
    const float s2 = __shfl(h[2], m);
    const float s3 = __shfl(h[3], m);
    const float s6 = __shfl(h[6], m);
    const float s7 = __shfl(h[7], m);
    v2f b0, b1;
    b0.x = hi ? s2 : h[0];  b0.y = hi ? s3 : h[1];
    b1.x = hi ? s6 : h[4];  b1.y = hi ? s7 : h[5];

    v8f acc = wmma4(a0, b0, c);
    acc = wmma4(a1, b1, acc);

#pragma unroll
    for (int r = 0; r < 8; ++r) h[r] = fmaxf(acc[r], 0.f);

    // store h_t: lane (hi==0) holds 8 consecutive output floats
    if (hi == 0) {
      float* p = hbuf + base + (long)t * 16;
      float4 o0 = {h[0], h[1], h[2], h[3]};
      float4 o1 = {h[4], h[5], h[6], h[7]};
      *(float4*)(p)     = o0;
      *(float4*)(p + 4) = o1;
    }
    xa = nxa; xb = nxb;
  }
}

// -------------------------------------------------------------------------
// Kernel 3: FF head. D1 = leaky_relu(h_tile(16x16) @ w0^T + b0) via 4 WMMAs,
// then out[row] = sum_n D1[row][n]*w1[n] + b1 via butterfly shfl reduction.
// -------------------------------------------------------------------------
__global__ void ff_kernel(const float* __restrict__ hbuf, // [Rn,16]
                          const float* __restrict__ w0,   // [16,16]
                          const float* __restrict__ b0v,  // [16]
                          const float* __restrict__ w1,   // [16]
                          const float* __restrict__ b1v,  // [1]
                          float* __restrict__ out) {      // [Rn]
  const int lane = threadIdx.x & 31;
  const int wave = blockIdx.x * (blockDim.x >> 5) + (threadIdx.x >> 5);
  const int m  = lane & 15;
  const int hi = lane >> 4;
  const long row0 = (long)wave * 16;

  // B = w0^T: B[k][n] = w0[n][k]; lane owns column n = m
  v2f bw[4];
#pragma unroll
  for (int ks = 0; ks < 4; ++ks) {
    const int k = ks * 4 + hi * 2;
    bw[ks].x = w0[m * 16 + k];
    bw[ks].y = w0[m * 16 + k + 1];
  }
  v8f acc;
  const float bias = b0v[m];
#pragma unroll
  for (int r = 0; r < 8; ++r) acc[r] = bias;

  const float* hr = hbuf + row0 * 16;
#pragma unroll
  for (int ks = 0; ks < 4; ++ks) {
    const int k = ks * 4 + hi * 2;
    v2f a;
    a.x = hr[m * 16 + k];
    a.y = hr[m * 16 + k + 1];
    acc = wmma4(a, bw[ks], acc);
  }

  // leaky_relu(x) = max(x, 0.01*x)
#pragma unroll
  for (int r = 0; r < 8; ++r) acc[r] = fmaxf(acc[r], 0.01f * acc[r]);

  // Second linear: per D row, dot with w1 across the 16 lanes of each half.
  const float w1v = w1[m];
  const float b1s = b1v[0];
#pragma unroll
  for (int r = 0; r < 8; ++r) {
    float p = acc[r] * w1v;
    p += __shfl_xor(p, 1);
    p += __shfl_xor(p, 2);
    p += __shfl_xor(p, 4);
    p += __shfl_xor(p, 8);
    if (m == 0) out[row0 + r + hi * 8] = p + b1s;
  }
}

extern "C" void kernel_launch(void* const* d_in, const int* in_sizes, int n_in,
                              void* d_out, int out_size, void* d_ws, size_t ws_size,
                              hipStream_t stream) {
  const float* x      = (const float*)d_in[0];
  const float* w_ih_f = (const float*)d_in[1];
  const float* w_hh_f = (const float*)d_in[2];
  const float* b_f    = (const float*)d_in[3];
  const float* w_ih_b = (const float*)d_in[4];
  const float* w_hh_b = (const float*)d_in[5];
  const float* b_b    = (const float*)d_in[6];
  const float* w0     = (const float*)d_in[7];
  const float* b0     = (const float*)d_in[8];
  const float* w1     = (const float*)d_in[9];
  const float* b1     = (const float*)d_in[10];
  float* out = (float*)d_out;

  float* xp   = (float*)d_ws;                  // Rn*16 f32 = 32 MB
  float* hbuf = xp + (size_t)Rn * 16;          // Rn*16 f32 = 32 MB

  // 32768 row-tiles of 16, 8 waves (256 threads) per block
  proj_kernel<<<Rn / 16 / 8, 256, 0, stream>>>(x, w_ih_f, w_ih_b, b_f, b_b, xp);
  // 2 directions x 8 batch-tiles, one wave each
  scan_kernel<<<16, 32, 0, stream>>>(xp, w_hh_f, w_hh_b, hbuf);
  ff_kernel<<<Rn / 16 / 8, 256, 0, stream>>>(hbuf, w0, b0, w1, b1, out);
}